// S4Layer_36017595744646
// MI455X (gfx1250) — compile-verified
//
#include <hip/hip_runtime.h>

// ---------------------------------------------------------------------------
// S4 linear state-space layer, CDNA5 (gfx1250) WMMA implementation.
//
// state_t = state_{t-1} @ A^T + u_t @ B^T      (state: 16 x 64)
// y_t     = state_t @ C^T + D * u_t            (y: 16 x 256)
//
// Pipeline (all f32, V_WMMA_F32_16X16X4_F32):
//  K1 : V[t] = u_t @ B^T          parallel GEMM, 4 n-tiles/wave (x read once)
//  K2a: Pow[j] = (A^T)^(j+1)      log-doubling, depth ~7
//  K2b: local chunk scans         64 blocks, ping-pong LDS, 4 WMMA chains
//  K2c: carry combine             1 block, 64 steps
//  K2d: fix-up state_t = l_t + carry @ Pow[t-t0]   (parallel, 1 block / t)
//  K3 : y = S @ C^T + D*u         parallel GEMM, 8 j-tiles/wave, f4 epilogue
//
// State buffer VL layout: VL[(t*64 + n)*16 + b]   (C/D frags -> float4 pairs)
// Pow layout: elem (j,k,n) at j*4096 + (k>>1)*128 + n*2 + (k&1)
//             (B-frag row pairs (k,k+1) are contiguous -> b64 loads)
// ---------------------------------------------------------------------------

typedef float v2f __attribute__((ext_vector_type(2)));
typedef float v4f __attribute__((ext_vector_type(4)));
typedef float v8f __attribute__((ext_vector_type(8)));

#define BATCH   16
#define DMODEL  256
#define DSTATE  64
#define SEQ     8192
#define LC      128          // chunk length
#define NCHUNK  (SEQ / LC)   // 64

__device__ __forceinline__ v8f wmma4(v2f a, v2f b, v8f c) {
  return __builtin_amdgcn_wmma_f32_16x16x4_f32(
      false, a, false, b, (short)0, c, false, false);
}

__device__ __forceinline__ size_t pw_idx(int j, int k, int n) {
  return (size_t)j * 4096 + (size_t)((k >> 1) * 128 + n * 2 + (k & 1));
}

// ---------------------------------------------------------------------------
// K1: V[t][n][b] = sum_m x[b][m][t] * B[n][m].  Grid (SEQ/16, BATCH), block 32.
// One wave computes the full N=64 strip -> x is read exactly once overall.
// ---------------------------------------------------------------------------
__global__ void k1_input_proj(const float* __restrict__ x,
                              const float* __restrict__ Bw,
                              float* __restrict__ VL) {
  const int lane = threadIdx.x & 31;
  const int lm = lane & 15, hf = lane >> 4;
  const int t0 = blockIdx.x * 16;
  const int b  = blockIdx.y;
  const float* X = x + (size_t)b * DMODEL * SEQ;

  v8f acc[4] = {};
  #pragma unroll 4
  for (int kt = 0; kt < DMODEL / 4; ++kt) {
    const int k0 = kt * 4 + 2 * hf;
    v2f a;
    a.x = X[(size_t)(k0 + 0) * SEQ + t0 + lm];
    a.y = X[(size_t)(k0 + 1) * SEQ + t0 + lm];
    #pragma unroll
    for (int nt = 0; nt < 4; ++nt) {
      const v2f bf = *(const v2f*)&Bw[(nt * 16 + lm) * DMODEL + k0];
      acc[nt] = wmma4(a, bf, acc[nt]);
    }
  }
  #pragma unroll
  for (int nt = 0; nt < 4; ++nt)
    #pragma unroll
    for (int r = 0; r < 8; ++r) {
      const int t = t0 + r + 8 * hf;
      VL[((size_t)t * DSTATE + nt * 16 + lm) * BATCH + b] = acc[nt][r];
    }
}

// ---------------------------------------------------------------------------
// K2a: powers of T (T[k][n] = A[n][k]) by log-doubling.
// Pow[j] = T^(j+1).  Level: Pow[have + i] = Pow[i] @ Pow[have-1], i<cnt.
// 1 block, 1024 threads (32 waves).  Depth = ceil(log2(LC)) levels.
// ---------------------------------------------------------------------------
__global__ void k2a_powers(const float* __restrict__ Aw,
                           float* __restrict__ Pow) {
  const int tid  = threadIdx.x;
  const int lane = tid & 31;
  const int lm = lane & 15, hf = lane >> 4;
  const int w  = tid >> 5;

  for (int idx = tid; idx < DSTATE * DSTATE; idx += 1024) {
    const int k = idx >> 6, n = idx & 63;
    Pow[pw_idx(0, k, n)] = Aw[n * DSTATE + k];
  }
  __threadfence();
  __syncthreads();

  int have = 1;
  while (have < LC) {
    const int cnt = (have < LC - have) ? have : (LC - have);
    const float* M = Pow + (size_t)(have - 1) * 4096;
    const int tasks = cnt * 16;
    for (int task = w; task < tasks; task += 32) {
      const int jn = have + (task >> 4);
      const int m0 = ((task >> 2) & 3) * 16;
      const int n0 = (task & 3) * 16;
      const float* As = Pow + (size_t)(jn - have) * 4096;
      const int row = m0 + lm;
      v8f acc = {};
      #pragma unroll
      for (int kt = 0; kt < 16; ++kt) {
        const int q = kt * 4 + 2 * hf;
        v2f a;
        a.x = As[(row >> 1) * 128 + q * 2 + (row & 1)];
        a.y = As[(row >> 1) * 128 + q * 2 + 2 + (row & 1)];
        const v2f bf = *(const v2f*)&M[(q >> 1) * 128 + (n0 + lm) * 2];
        acc = wmma4(a, bf, acc);
      }
      #pragma unroll
      for (int r = 0; r < 8; ++r) {
        const int rr = m0 + r + 8 * hf;
        Pow[pw_idx(jn, rr, n0 + lm)] = acc[r];
      }
    }
    __threadfence();
    __syncthreads();
    have += cnt;
  }
}

// ---------------------------------------------------------------------------
// K2b: local scans (zero init), in-place over VL.  Grid NCHUNK, block 128.
// Ping-pong LDS state, 4 independent WMMA accumulation chains per step,
// v_t folded in as a post-add (its load hides under the WMMA chain).
// ---------------------------------------------------------------------------
__global__ void k2b_local_scan(const float* __restrict__ Aw,
                               float* __restrict__ VL) {
  __shared__ float S[2][BATCH * DSTATE];
  const int tid  = threadIdx.x;
  const int lane = tid & 31;
  const int lm = lane & 15, hf = lane >> 4;
  const int n0 = (tid >> 5) * 16;

  for (int idx = tid; idx < BATCH * DSTATE; idx += 128) S[0][idx] = 0.0f;

  v2f bf[16];                       // T fragments for this wave's n-tile
  #pragma unroll
  for (int kt = 0; kt < 16; ++kt)
    bf[kt] = *(const v2f*)&Aw[(n0 + lm) * DSTATE + kt * 4 + 2 * hf];
  __syncthreads();

  int p = 0;
  const int t0 = blockIdx.x * LC;
  for (int tt = 0; tt < LC; ++tt) {
    const size_t vb = ((size_t)(t0 + tt) * DSTATE + n0 + lm) * BATCH + 8 * hf;
    const v4f va0 = *(const v4f*)&VL[vb];       // v_t, batch rows 8hf..8hf+3
    const v4f va1 = *(const v4f*)&VL[vb + 4];   // v_t, batch rows +4..+7

    v8f c[4] = {};
    #pragma unroll
    for (int kt = 0; kt < 16; ++kt) {
      const v2f a = *(const v2f*)&S[p][lm * DSTATE + kt * 4 + 2 * hf];
      c[kt & 3] = wmma4(a, bf[kt], c[kt & 3]);
    }
    const v8f s = (c[0] + c[1]) + (c[2] + c[3]);

    v4f o0, o1;
    #pragma unroll
    for (int i = 0; i < 4; ++i) { o0[i] = s[i] + va0[i]; o1[i] = s[i + 4] + va1[i]; }
    #pragma unroll
    for (int r = 0; r < 4; ++r) {
      S[1 - p][(r + 8 * hf) * DSTATE + n0 + lm]     = o0[r];
      S[1 - p][(r + 4 + 8 * hf) * DSTATE + n0 + lm] = o1[r];
    }
    *(v4f*)&VL[vb]     = o0;
    *(v4f*)&VL[vb + 4] = o1;
    __syncthreads();
    p ^= 1;
  }
}

// ---------------------------------------------------------------------------
// K2c: carries[g][b][n] = state entering chunk g.  1 block, 128 threads.
// carry_{g+1} = carry_g @ T^LC + l_end_g
// ---------------------------------------------------------------------------
__global__ void k2c_carries(const float* __restrict__ VL,
                            const float* __restrict__ Pow,
                            float* __restrict__ carries) {
  __shared__ float S[2][BATCH * DSTATE];
  const int tid  = threadIdx.x;
  const int lane = tid & 31;
  const int lm = lane & 15, hf = lane >> 4;
  const int n0 = (tid >> 5) * 16;

  for (int idx = tid; idx < BATCH * DSTATE; idx += 128) S[0][idx] = 0.0f;

  v2f pf[16];                       // fragments of T^LC = Pow[LC-1]
  #pragma unroll
  for (int kt = 0; kt < 16; ++kt) {
    const int q = kt * 4 + 2 * hf;
    pf[kt] = *(const v2f*)&Pow[(size_t)(LC - 1) * 4096 + (q >> 1) * 128 + (n0 + lm) * 2];
  }
  __syncthreads();

  int p = 0;
  for (int g = 0; g < NCHUNK; ++g) {
    #pragma unroll
    for (int r = 0; r < 8; ++r) {
      const int row = r + 8 * hf;
      carries[((size_t)g * BATCH + row) * DSTATE + n0 + lm] =
          S[p][row * DSTATE + n0 + lm];
    }
    const size_t eb =
        ((size_t)(g * LC + LC - 1) * DSTATE + n0 + lm) * BATCH + 8 * hf;
    const v4f la0 = *(const v4f*)&VL[eb];
    const v4f la1 = *(const v4f*)&VL[eb + 4];

    v8f c[4] = {};
    #pragma unroll
    for (int kt = 0; kt < 16; ++kt) {
      const v2f a = *(const v2f*)&S[p][lm * DSTATE + kt * 4 + 2 * hf];
      c[kt & 3] = wmma4(a, pf[kt], c[kt & 3]);
    }
    const v8f s = (c[0] + c[1]) + (c[2] + c[3]);
    #pragma unroll
    for (int r = 0; r < 4; ++r) {
      S[1 - p][(r + 8 * hf) * DSTATE + n0 + lm]     = s[r] + la0[r];
      S[1 - p][(r + 4 + 8 * hf) * DSTATE + n0 + lm] = s[r + 4] + la1[r];
    }
    __syncthreads();
    p ^= 1;
  }
}

// ---------------------------------------------------------------------------
// K2d: fix-up.  state_t = l_t + carry_g @ Pow[t - t0_g]  (in-place over VL)
// Grid SEQ, block 128.
// ---------------------------------------------------------------------------
__global__ void k2d_fixup(const float* __restrict__ carries,
                          const float* __restrict__ Pow,
                          float* __restrict__ VL) {
  const int tid  = threadIdx.x;
  const int lane = tid & 31;
  const int lm = lane & 15, hf = lane >> 4;
  const int n0 = (tid >> 5) * 16;
  const int t  = blockIdx.x;
  const int g  = t >> 7;            // t / LC
  const int j  = t & (LC - 1);      // exponent index: Pow[j] = T^(j+1)

  const size_t vb = ((size_t)t * DSTATE + n0 + lm) * BATCH + 8 * hf;
  const v4f l0 = *(const v4f*)&VL[vb];
  const v4f l1 = *(const v4f*)&VL[vb + 4];
  const float* Pj = Pow + (size_t)j * 4096;

  v8f c[2] = {};
  #pragma unroll
  for (int kt = 0; kt < 16; ++kt) {
    const int q = kt * 4 + 2 * hf;
    const v2f a = *(const v2f*)&carries[((size_t)g * BATCH + lm) * DSTATE + q];
    const v2f bf = *(const v2f*)&Pj[(q >> 1) * 128 + (n0 + lm) * 2];
    c[kt & 1] = wmma4(a, bf, c[kt & 1]);
  }
  const v8f s = c[0] + c[1];
  v4f o0, o1;
  #pragma unroll
  for (int i = 0; i < 4; ++i) { o0[i] = s[i] + l0[i]; o1[i] = s[i + 4] + l1[i]; }
  *(v4f*)&VL[vb]     = o0;
  *(v4f*)&VL[vb + 4] = o1;
}

// ---------------------------------------------------------------------------
// K3: y[b][j][t] = sum_n S[t][n][b] * C[j][n] + D[j] * x[b][j][t]
// Grid (SEQ/16, 2, BATCH), block 32.  8 j-tiles per wave -> VL read only 2x.
// ---------------------------------------------------------------------------
__global__ void k3_output_proj(const float* __restrict__ VL,
                               const float* __restrict__ Cw,
                               const float* __restrict__ Dw,
                               const float* __restrict__ x,
                               float* __restrict__ out) {
  const int lane = threadIdx.x & 31;
  const int lm = lane & 15, hf = lane >> 4;
  const int t0  = blockIdx.x * 16;
  const int jt0 = blockIdx.y * 8;
  const int b   = blockIdx.z;

  v8f acc[8] = {};
  #pragma unroll
  for (int kt = 0; kt < 16; ++kt) {
    const int k0 = kt * 4 + 2 * hf;
    const size_t ab = ((size_t)(t0 + lm) * DSTATE + k0) * BATCH + b;
    v2f a;
    a.x = VL[ab];
    a.y = VL[ab + BATCH];
    #pragma unroll
    for (int u = 0; u < 8; ++u) {
      const v2f bf = *(const v2f*)&Cw[((jt0 + u) * 16 + lm) * DSTATE + k0];
      acc[u] = wmma4(a, bf, acc[u]);
    }
  }
  #pragma unroll
  for (int u = 0; u < 8; ++u) {
    const int jrow = (jt0 + u) * 16 + lm;
    const float Dj = Dw[jrow];
    const size_t idx = ((size_t)b * DMODEL + jrow) * SEQ + t0 + 8 * hf;
    const v4f x0 = *(const v4f*)&x[idx];
    const v4f x1 = *(const v4f*)&x[idx + 4];
    v4f o0, o1;
    #pragma unroll
    for (int i = 0; i < 4; ++i) {
      o0[i] = acc[u][i]     + Dj * x0[i];
      o1[i] = acc[u][i + 4] + Dj * x1[i];
    }
    *(v4f*)&out[idx]     = o0;
    *(v4f*)&out[idx + 4] = o1;
  }
}

// ---------------------------------------------------------------------------
extern "C" void kernel_launch(void* const* d_in, const int* in_sizes, int n_in,
                              void* d_out, int out_size, void* d_ws, size_t ws_size,
                              hipStream_t stream) {
  const float* x  = (const float*)d_in[0];   // (16, 256, 8192)
  const float* Aw = (const float*)d_in[1];   // (64, 64)
  const float* Bw = (const float*)d_in[2];   // (64, 256)
  const float* Cw = (const float*)d_in[3];   // (256, 64)
  const float* Dw = (const float*)d_in[4];   // (256,)
  float* out = (float*)d_out;

  float* VL      = (float*)d_ws;                       // 32 MB
  float* Pow     = VL + (size_t)SEQ * BATCH * DSTATE;  // 2 MB
  float* carries = Pow + (size_t)LC * 4096;            // 256 KB

  k1_input_proj<<<dim3(SEQ / 16, BATCH), 32, 0, stream>>>(x, Bw, VL);
  k2a_powers<<<1, 1024, 0, stream>>>(Aw, Pow);
  k2b_local_scan<<<NCHUNK, 128, 0, stream>>>(Aw, VL);
  k2c_carries<<<1, 128, 0, stream>>>(VL, Pow, carries);
  k2d_fixup<<<SEQ, 128, 0, stream>>>(carries, Pow, VL);
  k3_output_proj<<<dim3(SEQ / 16, 2, BATCH), 32, 0, stream>>>(VL, Cw, Dw, x, out);
}